// VectorQuantizer_15899968930264
// MI455X (gfx1250) — compile-verified
//
#include <hip/hip_runtime.h>

typedef __attribute__((ext_vector_type(2))) float v2f;
typedef __attribute__((ext_vector_type(4))) float v4f;
typedef __attribute__((ext_vector_type(8))) float v8f;

#define NUM_E   1024
#define DIM     64
#define NVEC    32768      // 32*32*32 flattened z vectors
#define ZELEMS  2097152    // 32*64*32*32

// ---------------------------------------------------------------------------
// Kernel 0: en[j] = 0.5 * ||embedding[j]||^2
// ---------------------------------------------------------------------------
__global__ void vq_en_kernel(const float* __restrict__ emb, float* __restrict__ en) {
    int j = blockIdx.x * blockDim.x + threadIdx.x;
    if (j < NUM_E) {
        const v4f* row = (const v4f*)(emb + j * DIM);
        float s = 0.f;
#pragma unroll
        for (int t = 0; t < DIM / 4; ++t) {
            v4f v = row[t];
            s += v.x * v.x + v.y * v.y + v.z * v.z + v.w * v.w;
        }
        en[j] = 0.5f * s;
    }
}

// ---------------------------------------------------------------------------
// Kernel 1: per-row argmax of (z . e_j - 0.5||e_j||^2) via f32 WMMA.
// Block = 256 threads = 8 waves; wave handles 16 rows; block handles 128 rows.
// ---------------------------------------------------------------------------
__global__ void __launch_bounds__(256)
vq_argmax_kernel(const float* __restrict__ z, const float* __restrict__ emb,
                 const float* __restrict__ en, int* __restrict__ idx_out,
                 float* __restrict__ idxf_out) {
    __shared__ float s_en[NUM_E];           // 4 KB
    __shared__ float s_z[8 * 16 * 68];      // 34.8 KB, stride-68 pad (bank-conflict-free)

    const int tid  = threadIdx.x;
    const int wave = tid >> 5;
    const int lane = tid & 31;
    const int half = lane >> 4;             // 0: K even pair / rows 0-7 ; 1: K odd pair / rows 8-15
    const int l16  = lane & 15;

    for (int i = tid; i < NUM_E; i += 256) s_en[i] = en[i];

    const int row0 = blockIdx.x * 128 + wave * 16;
    // z_flat[n][d] lives at (n>>10)*65536 + d*1024 + (n&1023); 16 rows share one 1024-block.
    const int zbase = (row0 >> 10) * (DIM * 1024) + (row0 & 1023);
    float* zt = s_z + wave * (16 * 68);

    // Stage 16x64 z tile: lanes 0-15 load dim 2t, lanes 16-31 load dim 2t+1 (64B coalesced).
#pragma unroll
    for (int t = 0; t < 32; ++t) {
        int e = t * 32 + lane;
        int d = e >> 4;
        int r = e & 15;
        zt[r * 68 + d] = z[zbase + d * 1024 + r];
    }
    __syncthreads();

    // A fragments (16x4 f32 layout): lane<16 holds row l16, K={4t,4t+1}; lane>=16 K={4t+2,4t+3}.
    v2f a[16];
#pragma unroll
    for (int t = 0; t < 16; ++t)
        a[t] = *(const v2f*)(zt + l16 * 68 + 4 * t + 2 * half);

    float best_val[8];
    int   best_idx[8];
#pragma unroll
    for (int r = 0; r < 8; ++r) { best_val[r] = -3.4e38f; best_idx[r] = 0; }

    for (int jt = 0; jt < NUM_E / 16; ++jt) {
        const int   j  = jt * 16 + l16;         // code handled by this lane (C: N = lane%16)
        const float cj = s_en[j];
        if (jt + 1 < NUM_E / 16)
            __builtin_prefetch(emb + (j + 16) * DIM, 0, 1);   // global_prefetch_b8

        v8f acc;
#pragma unroll
        for (int r = 0; r < 8; ++r) acc[r] = -cj;             // fold -0.5||e||^2 into C

        // B fragments (4x16 f32): lane holds emb[code l16][K pair], mirrors A layout.
        const float* erow = emb + j * DIM + 2 * half;
#pragma unroll
        for (int t = 0; t < 16; ++t) {
            v2f b = *(const v2f*)(erow + 4 * t);
            acc = __builtin_amdgcn_wmma_f32_16x16x4_f32(
                false, a[t], false, b, (short)0, acc, false, false);
        }

#pragma unroll
        for (int r = 0; r < 8; ++r) {
            if (acc[r] > best_val[r]) { best_val[r] = acc[r]; best_idx[r] = j; }
        }
    }

    // Reduce across the 16 lanes that hold different codes for the same rows.
#pragma unroll
    for (int off = 8; off >= 1; off >>= 1) {
#pragma unroll
        for (int r = 0; r < 8; ++r) {
            float v = __shfl_xor(best_val[r], off, 16);
            int   i = __shfl_xor(best_idx[r], off, 16);
            if (v > best_val[r] || (v == best_val[r] && i < best_idx[r])) {
                best_val[r] = v; best_idx[r] = i;
            }
        }
    }

    if (l16 == 0) {                         // lane 0 -> rows 0-7, lane 16 -> rows 8-15
        const int rbase = row0 + half * 8;
#pragma unroll
        for (int r = 0; r < 8; ++r) {
            idx_out[rbase + r]  = best_idx[r];
            idxf_out[rbase + r] = (float)best_idx[r];
        }
    }
}

// ---------------------------------------------------------------------------
// Kernel 2: gather quantized vectors back to [B,D,H,W] + per-block SSE partials.
// grid 1024 x 256 threads, 8 elements per thread at stride 256 (fully coalesced).
// ---------------------------------------------------------------------------
__global__ void __launch_bounds__(256)
vq_gather_kernel(const float* __restrict__ z, const float* __restrict__ emb,
                 const int* __restrict__ idx, float* __restrict__ qout,
                 float* __restrict__ partial) {
    __shared__ float red[256];
    const int tid = threadIdx.x;
    float acc = 0.f;
    const int base = blockIdx.x * 2048;
#pragma unroll
    for (int k = 0; k < 8; ++k) {
        int o   = base + k * 256 + tid;     // flat index into [B,D,H,W]
        int b   = o >> 16;
        int d   = (o >> 10) & 63;
        int col = o & 1023;
        int n   = (b << 10) + col;
        float q  = emb[idx[n] * DIM + d];
        float zv = z[o];
        qout[o] = q;                        // quantized_st == quantized numerically
        float diff = zv - q;
        acc += diff * diff;
    }
    red[tid] = acc;
    __syncthreads();
    for (int s = 128; s > 0; s >>= 1) {
        if (tid < s) red[tid] += red[tid + s];
        __syncthreads();
    }
    if (tid == 0) partial[blockIdx.x] = red[0];
}

// ---------------------------------------------------------------------------
// Kernel 3: final deterministic reduction -> loss = 1.25 * mean((z-q)^2)
// ---------------------------------------------------------------------------
__global__ void vq_loss_kernel(const float* __restrict__ partial, float* __restrict__ loss) {
    __shared__ float red[256];
    const int tid = threadIdx.x;
    float a = 0.f;
    for (int i = tid; i < 1024; i += 256) a += partial[i];
    red[tid] = a;
    __syncthreads();
    for (int s = 128; s > 0; s >>= 1) {
        if (tid < s) red[tid] += red[tid + s];
        __syncthreads();
    }
    if (tid == 0) *loss = red[0] * (1.25f / (float)ZELEMS);
}

// ---------------------------------------------------------------------------
extern "C" void kernel_launch(void* const* d_in, const int* in_sizes, int n_in,
                              void* d_out, int out_size, void* d_ws, size_t ws_size,
                              hipStream_t stream) {
    const float* z   = (const float*)d_in[0];   // [32,64,32,32]
    const float* emb = (const float*)d_in[1];   // [1024,64]

    float* out  = (float*)d_out;
    float* qout = out;                  // 2,097,152 quantized_st
    float* loss = out + ZELEMS;         // 1 scalar
    float* idxf = out + ZELEMS + 1;     // 32,768 indices as float

    float* en      = (float*)d_ws;          // 1024 floats
    float* partial = en + NUM_E;            // 1024 floats
    int*   idx     = (int*)(partial + 1024); // 32768 ints

    vq_en_kernel    <<<NUM_E / 256, 256, 0, stream>>>(emb, en);
    vq_argmax_kernel<<<NVEC / 128,  256, 0, stream>>>(z, emb, en, idx, idxf);
    vq_gather_kernel<<<1024,        256, 0, stream>>>(z, emb, idx, qout, partial);
    vq_loss_kernel  <<<1,           256, 0, stream>>>(partial, loss);
}